// SpikingLayer_90958817395227
// MI455X (gfx1250) — compile-verified
//
#include <hip/hip_runtime.h>
#include <cstdint>

// IAF spiking-layer forward scan, MI455X (gfx1250).
// Bandwidth-bound streaming kernel: 256 MB HBM traffic, ~11 us floor @ 23.3 TB/s.
// CDNA5 async global->LDS loads (ASYNCcnt-tracked, NT-hinted, saddr form) give a
// deep prefetch pipeline (~32 MB in flight chip-wide); outputs use NT stores.

typedef float v4f __attribute__((ext_vector_type(4)));

namespace {

constexpr int kT        = 128;      // T_SIM
constexpr int kCHWShift = 15;       // CHW = 32*32*32 = 32768 = 1<<15
constexpr int kB        = 8;
constexpr int kTC       = 16;       // timesteps per pipeline chunk
constexpr int kNChunk   = kT / kTC; // 8
constexpr int kBlock    = 128;      // 4 waves/block, finer WGP distribution
constexpr int kGroups   = (kB << kCHWShift) / 4;  // 65536 float4 lanes of work

// CDNA5 async copy: global -> LDS, 128 bits per lane, ASYNCcnt-tracked.
// SADDR form: uniform 64-bit base (SGPR pair) + 32-bit per-lane byte offset.
// th:TH_LOAD_NT keeps the once-read input stream from thrashing L2.
__device__ __forceinline__ void async_ld_b128(uint32_t lds_addr, uint32_t gbyteoff,
                                              const float* sbase) {
  asm volatile("global_load_async_to_lds_b128 %0, %1, %2 th:TH_LOAD_NT"
               :: "v"(lds_addr), "v"(gbyteoff), "s"(sbase)
               : "memory");
}

__device__ __forceinline__ float spike1(float vmem, float& sub) {
  // threshold = 1, membrane_subtract = 1  ->  s = floor(v - sub) when >= 1
  float d = vmem - sub;
  float s = (d >= 1.0f) ? __builtin_floorf(d) : 0.0f;
  sub += s;
  return s;
}

} // namespace

__global__ __launch_bounds__(kBlock)
void iaf_scan_kernel(const float* __restrict__ in, float* __restrict__ out) {
  // Double-buffered staging: 2 x 16 timesteps x 128 lanes x 16B = 64 KB LDS.
  __shared__ v4f buf[2][kTC][kBlock];

  const int      tid  = threadIdx.x;
  const uint32_t g    = blockIdx.x * kBlock + tid;   // 0 .. 65535
  const uint32_t b    = g >> 13;                     // 8192 float4 groups per batch
  const uint32_t chw0 = (g & 8191u) << 2;            // 4 consecutive channels

  // Element index of (b, t=0, chw0); max byte offset 134 MB fits in u32.
  const uint32_t base     = ((b * kT) << kCHWShift) + chw0;
  float*         out_base = out + base;

  // Prologue: async-prefetch chunk 0 into buf[0].
#pragma unroll
  for (int tt = 0; tt < kTC; ++tt) {
    async_ld_b128((uint32_t)(uintptr_t)&buf[0][tt][tid],
                  (base + ((uint32_t)tt << kCHWShift)) << 2, in);
  }

  // Per-neuron running state (4 neurons per thread).
  float v0 = 0.f, v1 = 0.f, v2 = 0.f, v3 = 0.f;     // membrane cumsum
  float q0 = 0.f, q1 = 0.f, q2 = 0.f, q3 = 0.f;     // refractory subtraction

  for (int c = 0; c < kNChunk; ++c) {
    const int cur = c & 1;

    if (c + 1 < kNChunk) {
      // Issue next chunk's 16 async loads, then drain to <=16 outstanding:
      // in-order ASYNCcnt completion means the *current* chunk is then resident.
#pragma unroll
      for (int tt = 0; tt < kTC; ++tt) {
        const uint32_t t = (uint32_t)(c + 1) * kTC + tt;
        async_ld_b128((uint32_t)(uintptr_t)&buf[cur ^ 1][tt][tid],
                      (base + (t << kCHWShift)) << 2, in);
      }
      asm volatile("s_wait_asynccnt 16" ::: "memory");
    } else {
      asm volatile("s_wait_asynccnt 0" ::: "memory");
    }

    // Consume current chunk: sequential IAF scan, NT streaming stores.
#pragma unroll
    for (int tt = 0; tt < kTC; ++tt) {
      v4f x = buf[cur][tt][tid];          // ds_load_b128, own slot only
      v0 += x.x; v1 += x.y; v2 += x.z; v3 += x.w;

      v4f s;
      s.x = spike1(v0, q0);
      s.y = spike1(v1, q1);
      s.z = spike1(v2, q2);
      s.w = spike1(v3, q3);

      const uint32_t t = (uint32_t)c * kTC + tt;
      __builtin_nontemporal_store(s, (v4f*)(out_base + ((uint64_t)t << kCHWShift)));
    }
  }
}

extern "C" void kernel_launch(void* const* d_in, const int* in_sizes, int n_in,
                              void* d_out, int out_size, void* d_ws, size_t ws_size,
                              hipStream_t stream) {
  const float* in  = (const float*)d_in[0];   // [B, T, C, H, W] fp32, 0/1 spikes
  float*       out = (float*)d_out;           // same shape, fp32

  const dim3 grid(kGroups / kBlock);          // 512 blocks x 128 threads
  iaf_scan_kernel<<<grid, kBlock, 0, stream>>>(in, out);
}